// CandidateFinder_83571473645962
// MI455X (gfx1250) — compile-verified
//
#include <hip/hip_runtime.h>
#include <math.h>

#define LSH_BUCKETS 32
#define N_HASHES    4
#define K_MAX       64
#define D_DIM       64
#define L_SEQ       4096

typedef __attribute__((ext_vector_type(2))) float v2f;
typedef __attribute__((ext_vector_type(8))) float v8f;

// ---------------------------------------------------------------------------
// Kernel 1: binarize + LSH projection via V_WMMA_F32_16X16X4_F32, pack codes.
// One wave32 handles 16 tokens (M=16). D=64 is covered by 16 K-steps of K=4.
// A (16x4 f32, 2 VGPR/lane):  lanes 0-15: M=lane,   K = 4s+0, 4s+1
//                             lanes16-31: M=lane-16,K = 4s+2, 4s+3
// B (4x16 f32, 2 VGPR/lane):  N = lane%16; K rows split 0/2 and 1/3 halves.
// C/D (16x16 f32, 8 VGPR):    VGPR r -> M = r + 8*(lane/16), N = lane%16.
// Only N<4 columns are real hash projections; columns 4..15 are zero-padded.
// ---------------------------------------------------------------------------
__global__ void lsh_hash_wmma_kernel(const float* __restrict__ q,
                                     const float* __restrict__ k,
                                     const float* __restrict__ W,
                                     int* __restrict__ qcode,
                                     int* __restrict__ kcode,
                                     int BL) {
  const int lane    = threadIdx.x & 31;
  const int waveBlk = threadIdx.x >> 5;
  const int wave    = blockIdx.x * (blockDim.x >> 5) + waveBlk;
  const int tokBase = wave * 16;                 // 16 tokens per wave
  if (tokBase >= 2 * BL) return;                 // wave-uniform guard

  const int m   = lane & 15;                     // M within tile / N column
  const int hi  = lane >> 4;                     // half-wave selector
  const int tok = tokBase + m;
  const float* __restrict__ x =
      (tok < BL) ? (q + (size_t)tok * D_DIM) : (k + (size_t)(tok - BL) * D_DIM);

  const int   n    = m;                          // B/D column this lane feeds
  const float bmsk = (n < N_HASHES) ? 1.0f : 0.0f;
  const int   nn   = n & (N_HASHES - 1);         // clamped column (no branch)

  v8f c = {};
#pragma unroll
  for (int s = 0; s < 16; ++s) {
    const int k0 = 4 * s + 2 * hi;
    // A fragment: binary-quantized token values (exact in f32)
    v2f a;
    a.x = (x[k0]     > 0.0f) ? 1.0f : 0.0f;
    a.y = (x[k0 + 1] > 0.0f) ? 1.0f : 0.0f;
    // B fragment: W columns, zero-masked for n >= N_HASHES (no EXEC change)
    v2f b;
    b.x = W[k0 * N_HASHES + nn]       * bmsk;
    b.y = W[(k0 + 1) * N_HASHES + nn] * bmsk;
    c = __builtin_amdgcn_wmma_f32_16x16x4_f32(
        /*neg_a=*/false, a, /*neg_b=*/false, b,
        /*c_mod=*/(short)0, c, /*reuse_a=*/false, /*reuse_b=*/false);
  }

  // Per-lane hash contribution: h(proj) * 32^n  (n<4), else 0.
  // Python-style non-negative mod 32 == (& 31) in two's complement.
  int contrib[8];
#pragma unroll
  for (int r = 0; r < 8; ++r) {
    const int h = ((int)floorf(c[r] * 0.5f)) & (LSH_BUCKETS - 1);
    contrib[r] = (n < N_HASHES) ? (h << (5 * n)) : 0;
  }
  // Radix-sum the 4 hash columns per token: butterfly over lane groups {0..3},
  // {16..19}; lanes with n>=4 contribute zeros.
#pragma unroll
  for (int r = 0; r < 8; ++r) {
    int v = contrib[r];
    v += __shfl_xor(v, 1, 32);
    v += __shfl_xor(v, 2, 32);
    contrib[r] = v;
  }
  if (m == 0) {                                  // lanes 0 and 16 write codes
#pragma unroll
    for (int r = 0; r < 8; ++r) {
      const int t = tokBase + 8 * hi + r;
      if (t < BL) qcode[t]        = contrib[r];
      else        kcode[t - BL]   = contrib[r];
    }
  }
}

// ---------------------------------------------------------------------------
// Kernel 2: ordered first-K_MAX match scan. Key codes for one batch row are
// staged in LDS (16 KB << 320 KB/WGP); each wave32 owns one query row and
// walks j in 32-wide chunks using ballot + popcount-prefix to keep output
// slots in increasing-j order, with uniform early exit at 64 matches.
// ---------------------------------------------------------------------------
__global__ void lsh_match_kernel(const int* __restrict__ qcode,
                                 const int* __restrict__ kcode,
                                 int* __restrict__ out,
                                 int B, int L) {
  __shared__ int s_k[L_SEQ];
  const int lane          = threadIdx.x & 31;
  const int wave          = threadIdx.x >> 5;
  const int rowsPerBlock  = blockDim.x >> 5;
  const int blocksPerBat  = L / rowsPerBlock;
  const int b             = blockIdx.x / blocksPerBat;
  const int i             = (blockIdx.x % blocksPerBat) * rowsPerBlock + wave;

  for (int j = threadIdx.x; j < L; j += blockDim.x)
    s_k[j] = kcode[(size_t)b * L + j];
  __syncthreads();

  const int qc = qcode[(size_t)b * L + i];
  int* __restrict__ dst = out + ((size_t)b * L + i) * K_MAX;

  int count = 0;
  const unsigned lowMask = (1u << lane) - 1u;
  for (int chunk = 0; chunk < L; chunk += 32) {
    const int  j     = chunk + lane;
    const bool match = (s_k[j] == qc);
    const unsigned mask = __builtin_amdgcn_ballot_w32(match);
    if (match) {
      const int pos = count + __builtin_popcount(mask & lowMask);
      if (pos < K_MAX) dst[pos] = j;
    }
    count += __builtin_popcount(mask);           // wave-uniform
    if (count >= K_MAX) break;                   // uniform early exit
  }
  if (count > K_MAX) count = K_MAX;
  for (int slot = count + lane; slot < K_MAX; slot += 32)
    dst[slot] = -1;
}

extern "C" void kernel_launch(void* const* d_in, const int* in_sizes, int n_in,
                              void* d_out, int out_size, void* d_ws, size_t ws_size,
                              hipStream_t stream) {
  const float* query = (const float*)d_in[0];
  const float* key   = (const float*)d_in[1];
  const float* W     = (const float*)d_in[2];
  // d_in[3] = head_idx (unused by the reference output)

  const int L  = L_SEQ;
  const int BL = in_sizes[0] / D_DIM;            // B*L = 16384
  const int B  = BL / L;

  int* qcode = (int*)d_ws;                       // BL ints
  int* kcode = qcode + BL;                       // BL ints (128 KB total)
  int* out   = (int*)d_out;                      // (B, L, K_MAX) int32

  // Kernel 1: 2*BL tokens, 16 per wave, 8 waves (256 threads) per block.
  const int waves1  = (2 * BL) / 16;             // 2048
  const int blocks1 = waves1 / 8;                // 256
  lsh_hash_wmma_kernel<<<blocks1, 256, 0, stream>>>(query, key, W,
                                                    qcode, kcode, BL);

  // Kernel 2: 8 query rows per block (256 threads = 8 waves).
  const int rowsPerBlock = 8;
  const int blocks2 = B * (L / rowsPerBlock);    // 2048
  lsh_match_kernel<<<blocks2, rowsPerBlock * 32, 0, stream>>>(qcode, kcode,
                                                              out, B, L);
}